// PointRend_39779987096017
// MI455X (gfx1250) — compile-verified
//
#include <hip/hip_runtime.h>
#include <math.h>
#include <stdint.h>

typedef float v2f __attribute__((ext_vector_type(2)));
typedef float v8f __attribute__((ext_vector_type(8)));

#define NB 8      // batch
#define CL 2      // logit channels
#define CF 256    // feature channels
#define HF 256    // feat H
#define WF 256    // feat W
#define P  8192   // points per subdiv step
#define FCD 256   // hidden dim
#define KP 260    // padded MLP K (258 real + bias-ones row + zero pad)

// ---------------- weight prep: pack fc weights with bias folded into col 258 ----
__global__ __launch_bounds__(256)
void prep_w_k(const float* __restrict__ w0, const float* __restrict__ b0,
              const float* __restrict__ w1, const float* __restrict__ b1,
              const float* __restrict__ w2, const float* __restrict__ b2,
              float* __restrict__ Wp) {
  int i = blockIdx.x * 256 + threadIdx.x;
  int total = 3 * FCD * KP;
  if (i >= total) return;
  int l = i / (FCD * KP);
  int r = i % (FCD * KP);
  int o = r / KP, c = r % KP;
  const float* w  = (l == 0) ? w0 : (l == 1) ? w1 : w2;
  const float* bb = (l == 0) ? b0 : (l == 1) ? b1 : b2;
  float v = (c < 258) ? w[o * 258 + c] : ((c == 258) ? bb[o] : 0.0f);
  Wp[i] = v;
}

// ---------------- 2x bilinear upsample (half-pixel centers, clamped edges) -----
__global__ __launch_bounds__(256)
void upsample2x_k(const float* __restrict__ in, float* __restrict__ out, int Hin) {
  int W2 = Hin * 2;
  int total = NB * CL * W2 * W2;
  int i = blockIdx.x * 256 + threadIdx.x;
  if (i >= total) return;
  int x  = i % W2;
  int y  = (i / W2) % W2;
  int nc = i / (W2 * W2);
  float sx = ((float)x + 0.5f) * 0.5f - 0.5f;
  float sy = ((float)y + 0.5f) * 0.5f - 0.5f;
  float fx = floorf(sx), fy = floorf(sy);
  float wx = sx - fx, wy = sy - fy;
  int x0 = (int)fx, y0 = (int)fy;
  int x1 = x0 + 1, y1 = y0 + 1;
  x0 = x0 < 0 ? 0 : x0;  y0 = y0 < 0 ? 0 : y0;
  x1 = x1 > Hin - 1 ? Hin - 1 : x1;
  y1 = y1 > Hin - 1 ? Hin - 1 : y1;
  const float* p = in + (size_t)nc * Hin * Hin;
  float v00 = p[y0 * Hin + x0], v01 = p[y0 * Hin + x1];
  float v10 = p[y1 * Hin + x0], v11 = p[y1 * Hin + x1];
  out[i] = v00 * (1.f - wx) * (1.f - wy) + v01 * wx * (1.f - wy)
         + v10 * (1.f - wx) * wy        + v11 * wx * wy;
}

// ---------------- uncertainty -> monotone sortable uint keys -------------------
__global__ __launch_bounds__(256)
void unc_keys_k(const float* __restrict__ U, unsigned* __restrict__ keys, int HW) {
  int i = blockIdx.x * 256 + threadIdx.x;
  if (i >= NB * HW) return;
  int b = i / HW, pos = i % HW;
  const float* Ub = U + (size_t)b * CL * HW;
  float unc = -fabsf(Ub[pos] - Ub[HW + pos]);      // top2[1]-top2[0], C==2
  unsigned u = __float_as_uint(unc);
  u = (u & 0x80000000u) ? ~u : (u | 0x80000000u);  // larger float -> larger uint
  keys[i] = u;
}

// ---------------- exact Kth-largest via 4-round byte radix select --------------
__global__ __launch_bounds__(1024)
void radix_select_k(const unsigned* __restrict__ keys, int HW,
                    unsigned* __restrict__ thresh, unsigned* __restrict__ remE) {
  __shared__ unsigned hist[256];
  __shared__ unsigned s_prefix, s_rem;
  int b = blockIdx.x;
  const unsigned* kb = keys + (size_t)b * HW;
  int tid = threadIdx.x;
  if (tid == 0) { s_prefix = 0u; s_rem = (unsigned)P; }
  __syncthreads();
  for (int byte = 3; byte >= 0; --byte) {
    if (tid < 256) hist[tid] = 0u;
    __syncthreads();
    unsigned pre = s_prefix;
    int sh = byte * 8;
    for (int i = tid; i < HW; i += 1024) {
      unsigned k = kb[i];
      if (byte == 3 || (k >> (sh + 8)) == pre)
        atomicAdd(&hist[(k >> sh) & 255u], 1u);
    }
    __syncthreads();
    if (tid == 0) {
      unsigned rem = s_rem, cum = 0u;
      int v = 255;
      for (; v > 0; --v) {
        if (cum + hist[v] >= rem) break;
        cum += hist[v];
      }
      s_prefix = (pre << 8) | (unsigned)v;
      s_rem = rem - cum;
    }
    __syncthreads();
  }
  if (tid == 0) { thresh[b] = s_prefix; remE[b] = s_rem; }
}

// ---- deterministic compaction: all >thr, plus lowest-index remE equal to thr --
__global__ __launch_bounds__(1024)
void compact_topk_k(const unsigned* __restrict__ keys, int HW,
                    const unsigned* __restrict__ thresh,
                    const unsigned* __restrict__ remE,
                    int* __restrict__ sel) {
  __shared__ unsigned sg[1025];
  __shared__ unsigned se[1025];
  int b = blockIdx.x;
  const unsigned* kb = keys + (size_t)b * HW;
  int tid = threadIdx.x;
  int chunk = HW >> 10;              // HW is always a multiple of 1024
  int start = tid * chunk;
  unsigned thr = thresh[b];
  unsigned g = 0u, e = 0u;
  for (int i = 0; i < chunk; ++i) {
    unsigned k = kb[start + i];
    g += (k > thr);
    e += (k == thr);
  }
  sg[tid + 1] = g; se[tid + 1] = e;
  __syncthreads();
  if (tid == 0) {
    sg[0] = 0u; se[0] = 0u;
    for (int i = 1; i <= 1024; ++i) { sg[i] += sg[i - 1]; se[i] += se[i - 1]; }
  }
  __syncthreads();
  unsigned rem = remE[b];
  unsigned G = (unsigned)P - rem;    // exact count of keys > thr
  unsigned gpos = sg[tid], epos = se[tid];
  int* sb = sel + (size_t)b * P;
  for (int i = 0; i < chunk; ++i) {
    unsigned k = kb[start + i];
    if (k > thr) { sb[gpos++] = start + i; }
    else if (k == thr) { if (epos < rem) sb[G + epos] = start + i; epos++; }
  }
}

// ---- bilinear point-sample feat(256ch) + coarse(2ch); build padded X matrix ---
__global__ __launch_bounds__(256)
void sample_build_k(const int* __restrict__ sel, const float* __restrict__ feat,
                    const float* __restrict__ coarse,
                    float* __restrict__ Xa, float* __restrict__ Xb, int Wg) {
  int b = blockIdx.z;
  int j = blockIdx.x * 256 + threadIdx.x;
  int idx = sel[(size_t)b * P + j];
  float cx = ((float)(idx % Wg) + 0.5f) / (float)Wg;
  float cy = ((float)(idx / Wg) + 0.5f) / (float)Wg;   // H == W
  float* XA = Xa + (size_t)b * KP * P;
  float* XB = Xb + (size_t)b * KP * P;

  // ---- fine features (WFxHF grid), OOB taps contribute zero ----
  {
    float px = cx * (float)WF - 0.5f, py = cy * (float)HF - 0.5f;
    float fx = floorf(px), fy = floorf(py);
    float wx = px - fx, wy = py - fy;
    int x0 = (int)fx, y0 = (int)fy, x1 = x0 + 1, y1 = y0 + 1;
    int x0c = x0 < 0 ? 0 : (x0 > WF - 1 ? WF - 1 : x0);
    int x1c = x1 < 0 ? 0 : (x1 > WF - 1 ? WF - 1 : x1);
    int y0c = y0 < 0 ? 0 : (y0 > HF - 1 ? HF - 1 : y0);
    int y1c = y1 < 0 ? 0 : (y1 > HF - 1 ? HF - 1 : y1);
    float w00 = ((x0 >= 0) & (x0 < WF) & (y0 >= 0) & (y0 < HF)) ? (1.f - wx) * (1.f - wy) : 0.f;
    float w01 = ((x1 >= 0) & (x1 < WF) & (y0 >= 0) & (y0 < HF)) ? wx * (1.f - wy) : 0.f;
    float w10 = ((x0 >= 0) & (x0 < WF) & (y1 >= 0) & (y1 < HF)) ? (1.f - wx) * wy : 0.f;
    float w11 = ((x1 >= 0) & (x1 < WF) & (y1 >= 0) & (y1 < HF)) ? wx * wy : 0.f;
    int o00 = y0c * WF + x0c, o01 = y0c * WF + x1c;
    int o10 = y1c * WF + x0c, o11 = y1c * WF + x1c;
    const float* fb = feat + (size_t)b * CF * (HF * WF);
    for (int c = 0; c < CF; ++c) {
      const float* fc = fb + (size_t)c * (HF * WF);
      XA[(size_t)c * P + j] = w00 * fc[o00] + w01 * fc[o01] + w10 * fc[o10] + w11 * fc[o11];
    }
  }
  // ---- coarse logits (64x64 grid), 2 channels ----
  {
    const int Wc = 64;
    float px = cx * (float)Wc - 0.5f, py = cy * (float)Wc - 0.5f;
    float fx = floorf(px), fy = floorf(py);
    float wx = px - fx, wy = py - fy;
    int x0 = (int)fx, y0 = (int)fy, x1 = x0 + 1, y1 = y0 + 1;
    int x0c = x0 < 0 ? 0 : (x0 > Wc - 1 ? Wc - 1 : x0);
    int x1c = x1 < 0 ? 0 : (x1 > Wc - 1 ? Wc - 1 : x1);
    int y0c = y0 < 0 ? 0 : (y0 > Wc - 1 ? Wc - 1 : y0);
    int y1c = y1 < 0 ? 0 : (y1 > Wc - 1 ? Wc - 1 : y1);
    float w00 = ((x0 >= 0) & (x0 < Wc) & (y0 >= 0) & (y0 < Wc)) ? (1.f - wx) * (1.f - wy) : 0.f;
    float w01 = ((x1 >= 0) & (x1 < Wc) & (y0 >= 0) & (y0 < Wc)) ? wx * (1.f - wy) : 0.f;
    float w10 = ((x0 >= 0) & (x0 < Wc) & (y1 >= 0) & (y1 < Wc)) ? (1.f - wx) * wy : 0.f;
    float w11 = ((x1 >= 0) & (x1 < Wc) & (y1 >= 0) & (y1 < Wc)) ? wx * wy : 0.f;
    int o00 = y0c * Wc + x0c, o01 = y0c * Wc + x1c;
    int o10 = y1c * Wc + x0c, o11 = y1c * Wc + x1c;
    const float* cb = coarse + (size_t)b * CL * (Wc * Wc);
    for (int c = 0; c < CL; ++c) {
      const float* cc = cb + (size_t)c * (Wc * Wc);
      float v = w00 * cc[o00] + w01 * cc[o01] + w10 * cc[o10] + w11 * cc[o11];
      XA[(size_t)(CF + c) * P + j] = v;
      XB[(size_t)(CF + c) * P + j] = v;
    }
  }
  XA[(size_t)258 * P + j] = 1.0f;  XB[(size_t)258 * P + j] = 1.0f;   // bias row
  XA[(size_t)259 * P + j] = 0.0f;  XB[(size_t)259 * P + j] = 0.0f;   // zero pad row
}

// ---- MLP layer: Y[0..255][p] = relu(Wp(256xKP) @ X(KPxP)), via f32 WMMA -------
// Fragment layouts per CDNA5 ISA 7.12.2: A/B element k -> half = k>>1 (lane>=16),
// vgpr = k&1; lane%16 = m (A) / n (B). D: m = vgpr + 8*(lane>>4), n = lane%16.
__global__ __launch_bounds__(256)
void mlp_gemm_k(const float* __restrict__ X, float* __restrict__ Y,
                const float* __restrict__ Wp) {
  int wv   = threadIdx.x >> 5;        // wave 0..7
  int lane = threadIdx.x & 31;
  int ml   = lane & 15;
  int kh   = lane >> 4;
  int mo   = blockIdx.x * 128 + wv * 16;   // M base (grid.x = 2 -> 256)
  int no   = blockIdx.y * 64;              // N base (grid.y = 128 -> 8192)
  const float* Xb = X + (size_t)blockIdx.z * KP * P;
  float*       Yb = Y + (size_t)blockIdx.z * KP * P;
  const float* Wr = Wp + (size_t)(mo + ml) * KP + 2 * kh;

  v8f acc0 = {}, acc1 = {}, acc2 = {}, acc3 = {};
  for (int ko = 0; ko < KP; ko += 4) {
    v2f a = *(const v2f*)(Wr + ko);                    // 8B aligned: indices even
    const float* xp = Xb + (size_t)(ko + 2 * kh) * P + no + ml;
    v2f b0, b1, b2, b3;
    b0.x = xp[0];  b0.y = xp[P];
    b1.x = xp[16]; b1.y = xp[P + 16];
    b2.x = xp[32]; b2.y = xp[P + 32];
    b3.x = xp[48]; b3.y = xp[P + 48];
    acc0 = __builtin_amdgcn_wmma_f32_16x16x4_f32(false, a, false, b0, (short)0, acc0, false, false);
    acc1 = __builtin_amdgcn_wmma_f32_16x16x4_f32(false, a, false, b1, (short)0, acc1, false, false);
    acc2 = __builtin_amdgcn_wmma_f32_16x16x4_f32(false, a, false, b2, (short)0, acc2, false, false);
    acc3 = __builtin_amdgcn_wmma_f32_16x16x4_f32(false, a, false, b3, (short)0, acc3, false, false);
  }
  int nb = no + ml;
#pragma unroll
  for (int v = 0; v < 8; ++v) {
    int m = mo + v + 8 * kh;
    float* yr = Yb + (size_t)m * P + nb;
    yr[0]  = fmaxf(acc0[v], 0.0f);
    yr[16] = fmaxf(acc1[v], 0.0f);
    yr[32] = fmaxf(acc2[v], 0.0f);
    yr[48] = fmaxf(acc3[v], 0.0f);
  }
}

// ---- pred head (2 outputs) + scatter into upsampled logits --------------------
__global__ __launch_bounds__(256)
void pred_scatter_k(const float* __restrict__ X, const float* __restrict__ pw,
                    const float* __restrict__ pb, const int* __restrict__ sel,
                    float* __restrict__ logits, int HW) {
  int b = blockIdx.z;
  int j = blockIdx.x * 256 + threadIdx.x;
  const float* Xb = X + (size_t)b * KP * P;
  float s0 = pb[0], s1 = pb[1];
  for (int c = 0; c < 258; ++c) {
    float xv = Xb[(size_t)c * P + j];
    s0 += pw[c] * xv;          // pred_w (2,258) row-major
    s1 += pw[258 + c] * xv;
  }
  int idx = sel[(size_t)b * P + j];
  logits[(size_t)b * CL * HW + idx]      = s0;
  logits[(size_t)b * CL * HW + HW + idx] = s1;
}

extern "C" void kernel_launch(void* const* d_in, const int* in_sizes, int n_in,
                              void* d_out, int out_size, void* d_ws, size_t ws_size,
                              hipStream_t stream) {
  (void)in_sizes; (void)n_in; (void)out_size; (void)ws_size;
  const float* coarse = (const float*)d_in[0];
  const float* feat   = (const float*)d_in[1];
  const float* fc0w   = (const float*)d_in[2];
  const float* fc0b   = (const float*)d_in[3];
  const float* fc1w   = (const float*)d_in[4];
  const float* fc1b   = (const float*)d_in[5];
  const float* fc2w   = (const float*)d_in[6];
  const float* fc2b   = (const float*)d_in[7];
  const float* predw  = (const float*)d_in[8];
  const float* predb  = (const float*)d_in[9];
  float* out = (float*)d_out;

  uint8_t* ws = (uint8_t*)d_ws;
  size_t off = 0;
  auto take = [&](size_t bytes) -> void* {
    off = (off + 255) & ~(size_t)255;
    void* p = ws + off;
    off += bytes;
    return p;
  };
  float*    buf1   = (float*)take(sizeof(float) * NB * CL * 128 * 128);
  float*    buf2   = (float*)take(sizeof(float) * NB * CL * 256 * 256);
  unsigned* keys   = (unsigned*)take(sizeof(unsigned) * (size_t)NB * 512 * 512);
  unsigned* thresh = (unsigned*)take(sizeof(unsigned) * NB);
  unsigned* remE   = (unsigned*)take(sizeof(unsigned) * NB);
  int*      sel    = (int*)take(sizeof(int) * (size_t)NB * P);
  float*    Wp     = (float*)take(sizeof(float) * 3 * FCD * KP);
  float*    Xa     = (float*)take(sizeof(float) * (size_t)NB * KP * P);
  float*    Xb     = (float*)take(sizeof(float) * (size_t)NB * KP * P);

  prep_w_k<<<(3 * FCD * KP + 255) / 256, 256, 0, stream>>>(fc0w, fc0b, fc1w, fc1b,
                                                           fc2w, fc2b, Wp);

  const float* src = coarse;
  int Hin = 64;
  float* dsts[3] = {buf1, buf2, out};
  for (int s = 0; s < 3; ++s) {
    float* dst = dsts[s];
    int H2 = Hin * 2, HW2 = H2 * H2;
    int totalU = NB * CL * HW2;
    upsample2x_k<<<(totalU + 255) / 256, 256, 0, stream>>>(src, dst, Hin);
    unc_keys_k<<<(NB * HW2 + 255) / 256, 256, 0, stream>>>(dst, keys, HW2);
    radix_select_k<<<NB, 1024, 0, stream>>>(keys, HW2, thresh, remE);
    compact_topk_k<<<NB, 1024, 0, stream>>>(keys, HW2, thresh, remE, sel);
    sample_build_k<<<dim3(P / 256, 1, NB), 256, 0, stream>>>(sel, feat, coarse, Xa, Xb, H2);
    mlp_gemm_k<<<dim3(FCD / 128, P / 64, NB), 256, 0, stream>>>(Xa, Xb, Wp);
    mlp_gemm_k<<<dim3(FCD / 128, P / 64, NB), 256, 0, stream>>>(Xb, Xa, Wp + FCD * KP);
    mlp_gemm_k<<<dim3(FCD / 128, P / 64, NB), 256, 0, stream>>>(Xa, Xb, Wp + 2 * FCD * KP);
    pred_scatter_k<<<dim3(P / 256, 1, NB), 256, 0, stream>>>(Xb, predw, predb, sel, dst, HW2);
    src = dst;
    Hin = H2;
  }
}